// HRN_63247688401549
// MI455X (gfx1250) — compile-verified
//
#include <hip/hip_runtime.h>
#include <math.h>
#include <stdint.h>

// ---------------------------------------------------------------------------
// HRN routing on MI455X (gfx1250, wave32).
// Roofline: per step, (16 x 12288) @ (12288 x 12288) fp32 streams 604 MB of
// Wt from HBM (Wt = 576 MiB > 192 MB L2) -> 16 steps ~ 9.7 GB @ 23.3 TB/s.
// AI = 8 FLOP/byte -> ~186 TFLOP/s needed: memory bound, so keep fp32 and use
// the matrix pipe's V_WMMA_F32_16X16X4_F32 (M=16 == batch). P (48 MB) and
// basis (16 MB) stay L2-resident. X k-panels are DMA'd into LDS with the
// Tensor Data Mover (double-buffered, TENSORcnt-tracked), using TDM padding
// to get a bank-conflict-free LDS pitch of 514 DWORDs.
// ---------------------------------------------------------------------------

typedef float v2f __attribute__((ext_vector_type(2)));
typedef float v8f __attribute__((ext_vector_type(8)));
typedef unsigned int u32x4 __attribute__((ext_vector_type(4)));
typedef int i32x4 __attribute__((ext_vector_type(4)));
typedef int i32x8 __attribute__((ext_vector_type(8)));

#define NU    32      // n_units
#define HF    1024    // HASH features
#define BKD   128     // BASIS dim
#define DD    12288   // D = C*H*W
#define BB    16      // batch
#define DEPTH 16

// D = A(16x4) * B(4x16) + C(16x16), all fp32, wave32.
static __device__ __forceinline__ v8f wmma4(v2f a, v2f b, v8f c) {
  return __builtin_amdgcn_wmma_f32_16x16x4_f32(
      /*neg_a=*/false, a, /*neg_b=*/false, b,
      /*c_mod=*/(short)0, c, /*reuse_a=*/false, /*reuse_b=*/false);
}

// A-matrix 16x4 fp32 layout (ISA 7.12.2): lanes 0-15 hold M=lane with
// {K=k0,k0+1}, lanes 16-31 hold M=lane-16 with {K=k0+2,k0+3}.
static __device__ __forceinline__ v2f ldA(const float* __restrict__ A,
                                          int lda, int k0) {
  int lane = threadIdx.x & 31;
  int m    = lane & 15;
  int kk   = k0 + ((lane >> 4) << 1);
  v2f r;
  r.x = A[(size_t)m * lda + kk];
  r.y = A[(size_t)m * lda + kk + 1];
  return r;
}

// B-matrix 4x16 fp32: VGPR0 rows {k0 (lanes 0-15), k0+2 (lanes 16-31)},
// VGPR1 rows {k0+1, k0+3}; N = n0 + lane%16.  B row-major with leading dim.
static __device__ __forceinline__ v2f ldB(const float* __restrict__ B,
                                          int ldb, int k0, int n0) {
  int lane = threadIdx.x & 31;
  int n    = n0 + (lane & 15);
  int kk   = k0 + ((lane >> 4) << 1);
  v2f r;
  r.x = B[(size_t)kk * ldb + n];
  r.y = B[(size_t)(kk + 1) * ldb + n];
  return r;
}

// Same but B stored transposed: B[k][n] = Bt[n*ldb + k]
static __device__ __forceinline__ v2f ldBt(const float* __restrict__ Bt,
                                           int ldb, int k0, int n0) {
  int lane = threadIdx.x & 31;
  int n    = n0 + (lane & 15);
  int kk   = k0 + ((lane >> 4) << 1);
  v2f r;
  r.x = Bt[(size_t)n * ldb + kk];
  r.y = Bt[(size_t)n * ldb + kk + 1];
  return r;
}

// C/D 16x16 fp32: VGPR j holds M=j (lanes 0-15) and M=j+8 (lanes 16-31).
static __device__ __forceinline__ void stC(float* __restrict__ C, int ldc,
                                           v8f c) {
  int lane = threadIdx.x & 31;
  int n    = lane & 15;
  int mb   = (lane >> 4) << 3;
#pragma unroll
  for (int j = 0; j < 8; ++j) C[(size_t)(mb + j) * ldc + n] = c[j];
}

// --------------------------------------------------------------------------
// coeff(b,u,k) = sum_f H(b,f) * basis(u,f,k).   one wave per (u, 16-col tile)
__global__ void k_coeff(const float* __restrict__ Hh,
                        const float* __restrict__ basis,
                        float* __restrict__ coeff) {
  int u = blockIdx.x >> 3;     // 32 units
  int t = blockIdx.x & 7;      // 8 tiles over BKD=128
  const float* bu = basis + (size_t)u * HF * BKD;
  v8f c = {};
  for (int k0 = 0; k0 < HF; k0 += 4) {
    v2f a = ldA(Hh, HF, k0);
    v2f b = ldB(bu, BKD, k0, t * 16);
    c = wmma4(a, b, c);
  }
  stC(coeff + (size_t)u * BKD + t * 16, NU * BKD, c);
}

// proj(b,u,f) = sum_k coeff(b,u,k) * basis(u,f,k).  basis accessed transposed.
__global__ void k_proj(const float* __restrict__ coeff,
                       const float* __restrict__ basis,
                       float* __restrict__ proj) {
  int u = blockIdx.x >> 6;     // 32 units
  int t = blockIdx.x & 63;     // 64 tiles over HF=1024
  const float* bu = basis + (size_t)u * HF * BKD;
  v8f c = {};
  for (int k0 = 0; k0 < BKD; k0 += 4) {
    v2f a = ldA(coeff + (size_t)u * BKD, NU * BKD, k0);
    v2f b = ldBt(bu, BKD, k0, t * 16);
    c = wmma4(a, b, c);
  }
  stC(proj + (size_t)u * HF + t * 16, NU * HF, c);
}

// norms -> masked argmax -> residual accumulate.  Single block, 512 threads.
__global__ void k_route(const float* __restrict__ proj,
                        const float* __restrict__ Hh,
                        float* __restrict__ outacc, int* __restrict__ mask,
                        int* __restrict__ route, int* __restrict__ sel,
                        int step) {
  __shared__ float mags[BB * NU];
  __shared__ int ssel[BB];
  int t = threadIdx.x;
  int b = t >> 5, u = t & 31;
  const float* pp = proj + (size_t)b * NU * HF + (size_t)u * HF;
  float s = 0.f;
  for (int f = 0; f < HF; ++f) { float v = pp[f]; s += v * v; }
  mags[t] = s;                              // mags^2 (argmax-equivalent)
  __syncthreads();
  if (t < BB) {
    float best = -1.f;
    int bi = 0;
    for (int u2 = 0; u2 < NU; ++u2) {
      if (mask[t * NU + u2]) {
        float m = mags[t * NU + u2];
        if (m > best) { best = m; bi = u2; }
      }
    }
    ssel[t] = bi;
    sel[t]  = bi;
    mask[t * NU + bi]       = 0;
    route[t * DEPTH + step] = bi;
  }
  __syncthreads();
  for (int i = t; i < BB * HF; i += 512) {
    int bb2 = i >> 10, ff = i & (HF - 1);
    outacc[i] += Hh[i] -
                 proj[(size_t)bb2 * NU * HF + (size_t)ssel[bb2] * HF + ff];
  }
}

// --------------------------------------------------------------------------
// TDM descriptor: load one 16 x PANEL fp32 tile of X into LDS, with 1 DWORD
// of LDS padding inserted every 256 DWORDs -> LDS row pitch 514 DWORDs.
#define PANEL 512
#define XP    514      // padded LDS pitch in DWORDs
#define NP    (DD / PANEL)

#if __has_builtin(__builtin_amdgcn_tensor_load_to_lds)
#define USE_TDM 1
static __device__ __forceinline__ void tdm_load_x(const float* gsrc,
                                                  uint32_t lds_off) {
  uint64_t ga = (uint64_t)(uintptr_t)gsrc;
  u32x4 g0;
  g0.x = 1u;                                        // count=1, user D#
  g0.y = lds_off;                                   // lds_addr (bytes)
  g0.z = (uint32_t)(ga & 0xFFFFFFFFu);              // global_addr[31:0]
  g0.w = (uint32_t)((ga >> 32) & 0x01FFFFFFu) | (2u << 30);  // [56:32]|type=2
  i32x8 g1;
  g1[0] = (int)((2u << 16) | (1u << 20) | (7u << 22));  // 4B elem, pad 1DW/256DW
  g1[1] = (int)((DD & 0xFFFFu) << 16);              // tensor_dim0[15:0]
  g1[2] = (int)(((DD >> 16) & 0xFFFFu) | (BB << 16));   // dim0 hi | dim1 lo
  g1[3] = (int)(PANEL << 16);                       // dim1 hi=0 | tile_dim0
  g1[4] = (int)(BB);                                // tile_dim1=16, tile_dim2=0
  g1[5] = (int)(DD);                                // tensor_dim0_stride lo
  g1[6] = 0;                                        // stride hi | dim1_stride lo
  g1[7] = 0;
  i32x4 gz4 = {0, 0, 0, 0};                         // dims 2..4 unused (2D)
  i32x8 gz8 = {0, 0, 0, 0, 0, 0, 0, 0};
  // 6-arg form (clang-23 / therock-10.0 headers): extra i32x8 group, zeroed.
  __builtin_amdgcn_tensor_load_to_lds(g0, g1, gz4, gz4, gz8, 0);
}
#endif

// A-fragment read from the TDM-padded LDS panel (pad fixup at column 256).
static __device__ __forceinline__ v2f ldA_pad(const float* __restrict__ Xs,
                                              int k0) {
  int lane = threadIdx.x & 31;
  int m    = lane & 15;
  int kk   = k0 + ((lane >> 4) << 1);
  const float* row = Xs + m * XP + (kk >= 256 ? 1 : 0);
  v2f r;
  r.x = row[kk];
  r.y = row[kk + 1];
  return r;
}

// Xn = tanh(X @ Wt + bias[sel]).  HBM-bound GEMM: 192 blocks x 4 waves, each
// wave owns a 16-column tile; X k-panels DMA'd to LDS via TDM, double-buffered
// (wave 0 issues panel p+1, s_wait_tensorcnt 1 guarantees panel p resident).
__global__ void k_xform(const float* __restrict__ X,
                        const float* __restrict__ Wt,
                        const float* __restrict__ bias,
                        const int* __restrict__ sel,
                        float* __restrict__ Xn) {
  __shared__ float Xs[2][BB * XP];
  int t    = threadIdx.x;
  int wave = t >> 5;
  int n0   = blockIdx.x * 64 + wave * 16;
  v8f c = {};
#if defined(USE_TDM)
  uint32_t lds0 = (uint32_t)(uintptr_t)(void*)&Xs[0][0];
  uint32_t lds1 = (uint32_t)(uintptr_t)(void*)&Xs[1][0];
  if (wave == 0) tdm_load_x(X, lds0);
  for (int pi = 0; pi < NP; ++pi) {
    if (wave == 0) {
      if (pi + 1 < NP) {
        tdm_load_x(X + (size_t)(pi + 1) * PANEL, (pi & 1) ? lds0 : lds1);
        __builtin_amdgcn_s_wait_tensorcnt(1);   // panel pi complete (in-order)
      } else {
        __builtin_amdgcn_s_wait_tensorcnt(0);
      }
    }
    __syncthreads();
    const float* xsb = Xs[pi & 1];
    int pk = pi * PANEL;
#pragma unroll 4
    for (int k0 = 0; k0 < PANEL; k0 += 4) {
      v2f a = ldA_pad(xsb, k0);
      v2f b = ldB(Wt, DD, pk + k0, n0);
      c = wmma4(a, b, c);
    }
    __syncthreads();   // panel buffer free for the next TDM overwrite
  }
#else
  for (int pi = 0; pi < NP; ++pi) {
    __syncthreads();
    for (int i = t; i < BB * PANEL; i += blockDim.x) {
      int bb2 = i >> 9, kk = i & (PANEL - 1);
      Xs[0][bb2 * XP + kk + (kk >= 256 ? 1 : 0)] =
          X[(size_t)bb2 * DD + pi * PANEL + kk];
    }
    __syncthreads();
    int pk = pi * PANEL;
#pragma unroll 4
    for (int k0 = 0; k0 < PANEL; k0 += 4) {
      v2f a = ldA_pad(Xs[0], k0);
      v2f b = ldB(Wt, DD, pk + k0, n0);
      c = wmma4(a, b, c);
    }
  }
#endif
  int lane = t & 31;
  int n    = n0 + (lane & 15);
  int mb   = (lane >> 4) << 3;
#pragma unroll
  for (int j = 0; j < 8; ++j) {
    int m   = mb + j;
    float v = c[j] + bias[(size_t)sel[m] * DD + n];
    Xn[(size_t)m * DD + n] = tanhf(v);
  }
}

// --------------------------------------------------------------------------
// H = X @ P, split-K 8 ways for parallelism (P L2-resident), deterministic
// fixed-order reduction (no fp atomics: graph replays must be bit-stable).
#define HKS 8
#define HKC (DD / HKS)   // 1536
__global__ void k_hash_part(const float* __restrict__ X,
                            const float* __restrict__ P,
                            float* __restrict__ part) {
  int n0 = blockIdx.x * 16;
  int ks = blockIdx.y;
  int kb = ks * HKC;
  v8f c = {};
  for (int k0 = 0; k0 < HKC; k0 += 4) {
    v2f a = ldA(X + kb, DD, k0);
    v2f b = ldB(P + (size_t)kb * HF, HF, k0, n0);
    c = wmma4(a, b, c);
  }
  stC(part + (size_t)ks * BB * HF + n0, HF, c);
}

__global__ void k_hash_reduce(const float* __restrict__ part,
                              float* __restrict__ Hh) {
  int i = blockIdx.x * 256 + threadIdx.x;
  float s = 0.f;
#pragma unroll
  for (int ks = 0; ks < HKS; ++ks) s += part[(size_t)ks * BB * HF + i];
  Hh[i] = s;
}

__global__ void k_init(const float* __restrict__ x, float* __restrict__ Xc,
                       float* __restrict__ outacc, int* __restrict__ mask) {
  int i = blockIdx.x * blockDim.x + threadIdx.x;
  if (i < BB * DD) Xc[i] = x[i];
  if (i < BB * HF) outacc[i] = 0.f;
  if (i < BB * NU) mask[i] = 1;
}

__global__ void k_final(const float* __restrict__ outacc,
                        const int* __restrict__ route,
                        float* __restrict__ dout) {
  int i = blockIdx.x * blockDim.x + threadIdx.x;
  if (i < BB * HF) dout[i] = outacc[i];
  if (i < BB * DEPTH) ((int*)dout)[BB * HF + i] = route[i];
}

extern "C" void kernel_launch(void* const* d_in, const int* in_sizes, int n_in,
                              void* d_out, int out_size, void* d_ws,
                              size_t ws_size, hipStream_t stream) {
  (void)in_sizes; (void)n_in; (void)out_size; (void)ws_size;
  const float* x     = (const float*)d_in[0];
  const float* P     = (const float*)d_in[1];
  const float* basis = (const float*)d_in[2];
  const float* Wt    = (const float*)d_in[3];
  const float* bias  = (const float*)d_in[4];

  char* ws = (char*)d_ws;
  size_t off = 0;
  auto alloc = [&](size_t bytes) {
    char* p = ws + off;
    off += (bytes + 255) & ~(size_t)255;
    return p;
  };
  float* Xa    = (float*)alloc(sizeof(float) * BB * DD);
  float* Xb    = (float*)alloc(sizeof(float) * BB * DD);
  float* Hh    = (float*)alloc(sizeof(float) * BB * HF);
  float* coeff = (float*)alloc(sizeof(float) * BB * NU * BKD);
  float* proj  = (float*)alloc(sizeof(float) * BB * NU * HF);
  float* outac = (float*)alloc(sizeof(float) * BB * HF);
  float* part  = (float*)alloc(sizeof(float) * HKS * BB * HF);
  int*   mask  = (int*)alloc(sizeof(int) * BB * NU);
  int*   route = (int*)alloc(sizeof(int) * BB * DEPTH);
  int*   sel   = (int*)alloc(sizeof(int) * BB);

  k_init<<<(BB * DD + 255) / 256, 256, 0, stream>>>(x, Xa, outac, mask);
  k_hash_part<<<dim3(HF / 16, HKS), 32, 0, stream>>>(Xa, P, part);
  k_hash_reduce<<<(BB * HF) / 256, 256, 0, stream>>>(part, Hh);

  float* Xc = Xa;
  float* Xn = Xb;
  for (int s = 0; s < DEPTH; ++s) {
    k_coeff<<<NU * (BKD / 16), 32, 0, stream>>>(Hh, basis, coeff);
    k_proj<<<NU * (HF / 16), 32, 0, stream>>>(coeff, basis, proj);
    k_route<<<1, 512, 0, stream>>>(proj, Hh, outac, mask, route, sel, s);
    k_xform<<<DD / 64, 128, 0, stream>>>(Xc, Wt, bias, sel, Xn);
    k_hash_part<<<dim3(HF / 16, HKS), 32, 0, stream>>>(Xn, P, part);
    k_hash_reduce<<<(BB * HF) / 256, 256, 0, stream>>>(part, Hh);
    float* tmp = Xc; Xc = Xn; Xn = tmp;
  }
  k_final<<<(BB * HF + 255) / 256, 256, 0, stream>>>(outac, route,
                                                     (float*)d_out);
}